// ColorHead_46428596470407
// MI455X (gfx1250) — compile-verified
//
#include <hip/hip_runtime.h>

// ---------------------------------------------------------------------------
// ColorHead pipeline for MI455X (gfx1250, wave32).
// B=8, C=3, H=W=512, half=16. Bandwidth-bound; WMMA f32 16x16x4 for 1x1 convs.
// ---------------------------------------------------------------------------

#define WIDTH 512
#define HW    262144        // 512*512
#define NB    8
#define NPIXF 2097152.0f    // B*H*W per-channel BN sample count

typedef __attribute__((ext_vector_type(2))) float v2f;
typedef __attribute__((ext_vector_type(8))) float v8f;

__device__ __forceinline__ float lrelu(float v) { return v >= 0.0f ? v : 0.01f * v; }

// ---------------------------------------------------------------------------
// Kernel 1: quantize + 11x11 reflect-padded sliding-window mode -> xq = mode/16
// Block = 16x16 output tile (256 threads), LDS holds 26x26 bin-index tile.
// Packed histogram: 17 bins, 8 bits each (max count 121), 5 u32 registers.
// ---------------------------------------------------------------------------
__global__ void k_mode(const float* __restrict__ x, float* __restrict__ xq) {
  __shared__ int tile[26 * 26];
  const int plane = blockIdx.z;                 // b*3 + c
  const int oy0 = blockIdx.y * 16, ox0 = blockIdx.x * 16;
  const float* xp = x + (size_t)plane * HW;
  const int tid = threadIdx.x;

  for (int i = tid; i < 26 * 26; i += 256) {
    int ly = i / 26, lx = i - ly * 26;
    int gy = oy0 - 5 + ly;
    gy = gy < 0 ? -gy : (gy >= WIDTH ? 2 * WIDTH - 2 - gy : gy);   // reflect
    int gx = ox0 - 5 + lx;
    gx = gx < 0 ? -gx : (gx >= WIDTH ? 2 * WIDTH - 2 - gx : gx);
    tile[i] = __float2int_rn(xp[gy * WIDTH + gx] * (255.0f / 16.0f));
  }
  __syncthreads();

  const int ty = tid >> 4, tx = tid & 15;
  unsigned cnt[5] = {0u, 0u, 0u, 0u, 0u};
#pragma unroll
  for (int dy = 0; dy < 11; ++dy) {
#pragma unroll
    for (int dx = 0; dx < 11; ++dx) {
      int bin = tile[(ty + dy) * 26 + tx + dx];
      cnt[bin >> 2] += 1u << ((bin & 3) * 8);
    }
  }
  int best = -1, bb = 0;
#pragma unroll
  for (int b = 0; b < 17; ++b) {
    int c = (int)((cnt[b >> 2] >> ((b & 3) * 8)) & 255u);
    if (c > best) { best = c; bb = b; }           // first max -> smallest bin
  }
  xq[(size_t)plane * HW + (oy0 + ty) * WIDTH + ox0 + tx] = (float)bb * 0.0625f;
}

// ---------------------------------------------------------------------------
// Kernel 2: conv1x1 (3->16) + bias via WMMA f32 16x16x4; emit pre-BN h1 and
// per-channel sum/sumsq (register -> LDS atomics -> global atomics).
// Wave handles 8 tiles of 16 pixels; block (8 waves) covers 1024 pixels.
// A[m][k] = pw1[m][k] (k<3), A[m][3] = pb1[m]; B[3][n] = 1 -> bias folded in.
// All channel loads issued unconditionally; half-wave split done with value
// selects (v_cndmask), so no exec-predicated loads around the WMMAs.
// ---------------------------------------------------------------------------
__global__ void k_conv1(const float* __restrict__ xq, const float* __restrict__ pw1,
                        const float* __restrict__ pb1, float* __restrict__ h1,
                        float* __restrict__ st) {
  __shared__ float lsum[16], lsq[16];
  const int tid = threadIdx.x;
  const int lane = tid & 31, wave = tid >> 5;
  const int n = lane & 15, hi = lane >> 4;
  if (tid < 16) { lsum[tid] = 0.0f; lsq[tid] = 0.0f; }
  __syncthreads();

  const int G0 = blockIdx.x * 1024;
  const int b = G0 >> 18;
  const int pbase = (G0 & (HW - 1)) + wave * 128;
  const float* xb = xq + (size_t)b * 3 * HW;
  float* hb = h1 + (size_t)b * 16 * HW;

  v2f a;                                          // A 16x4: lane m, halves K{0,1}|{2,3}
  a.x = hi ? pw1[n * 3 + 2] : pw1[n * 3 + 0];
  a.y = hi ? pb1[n]         : pw1[n * 3 + 1];

  float ls[8], lq[8];
#pragma unroll
  for (int r = 0; r < 8; ++r) { ls[r] = 0.0f; lq[r] = 0.0f; }

#pragma unroll
  for (int t = 0; t < 8; ++t) {
    const int p = pbase + t * 16 + n;
    float c0 = xb[p];                             // unconditional, coalesced
    float c1 = xb[HW + p];
    float c2 = xb[2 * HW + p];
    v2f bm;                                       // B 4x16: lane n, halves K{0,1}|{2,3}
    bm.x = hi ? c2   : c0;
    bm.y = hi ? 1.0f : c1;
    v8f c = {0.f, 0.f, 0.f, 0.f, 0.f, 0.f, 0.f, 0.f};
    c = __builtin_amdgcn_wmma_f32_16x16x4_f32(false, a, false, bm, (short)0, c, false, false);
#pragma unroll
    for (int r = 0; r < 8; ++r) {                 // D: VGPR r -> channel r + 8*hi
      float v = c[r];
      hb[(size_t)(r + 8 * hi) * HW + p] = v;
      ls[r] += v; lq[r] += v * v;
    }
  }
#pragma unroll
  for (int r = 0; r < 8; ++r) {
    atomicAdd(&lsum[r + 8 * hi], ls[r]);
    atomicAdd(&lsq[r + 8 * hi], lq[r]);
  }
  __syncthreads();
  if (tid < 16) { atomicAdd(&st[tid], lsum[tid]); atomicAdd(&st[16 + tid], lsq[tid]); }
}

// ---------------------------------------------------------------------------
// Kernel 3: BN finalize -> scale/shift.  mean=sum/N, var=sumsq/N-mean^2 (biased)
// gmod lets 6x32 iteration channels share the 32-entry gamma/beta.
// ---------------------------------------------------------------------------
__global__ void k_finalize(const float* __restrict__ st, const float* __restrict__ gamma,
                           const float* __restrict__ beta, float* __restrict__ scale,
                           float* __restrict__ shift, int nch, int gmod) {
  int c = threadIdx.x;
  if (c >= nch) return;
  float mean = st[c] / NPIXF;
  float var = st[nch + c] / NPIXF - mean * mean;
  var = fmaxf(var, 0.0f);
  float inv = rsqrtf(var + 1e-5f);
  float s = gamma[c % gmod] * inv;
  scale[c] = s;
  shift[c] = beta[c % gmod] - mean * s;
}

// ---------------------------------------------------------------------------
// Kernel 4: apply BN1+LReLU on load, conv1x1 (16->16) via 4 chained WMMAs,
// add bias, store IN PLACE (pre-BN2), accumulate BN2 stats.
// ---------------------------------------------------------------------------
__global__ void k_conv2(float* __restrict__ h1, const float* __restrict__ pw2,
                        const float* __restrict__ pb2, const float* __restrict__ aff1,
                        float* __restrict__ st) {
  __shared__ float lsum[16], lsq[16];
  const int tid = threadIdx.x;
  const int lane = tid & 31, wave = tid >> 5;
  const int n = lane & 15, hi = lane >> 4;
  if (tid < 16) { lsum[tid] = 0.0f; lsq[tid] = 0.0f; }
  __syncthreads();

  const int G0 = blockIdx.x * 1024;
  const int b = G0 >> 18;
  const int pbase = (G0 & (HW - 1)) + wave * 128;
  float* hb = h1 + (size_t)b * 16 * HW;

  v2f A4[4];
#pragma unroll
  for (int k = 0; k < 4; ++k) {
    A4[k].x = pw2[n * 16 + 4 * k + 2 * hi];
    A4[k].y = pw2[n * 16 + 4 * k + 2 * hi + 1];
  }

  float ls[8], lq[8];
#pragma unroll
  for (int r = 0; r < 8; ++r) { ls[r] = 0.0f; lq[r] = 0.0f; }

#pragma unroll 2
  for (int t = 0; t < 8; ++t) {
    const int p = pbase + t * 16 + n;
    float inv[16];
#pragma unroll
    for (int cc = 0; cc < 16; ++cc)
      inv[cc] = lrelu(hb[(size_t)cc * HW + p] * aff1[cc] + aff1[16 + cc]);

    v8f c = {0.f, 0.f, 0.f, 0.f, 0.f, 0.f, 0.f, 0.f};
#pragma unroll
    for (int k = 0; k < 4; ++k) {
      v2f bm;
      bm.x = inv[4 * k + 2 * hi];
      bm.y = inv[4 * k + 2 * hi + 1];
      c = __builtin_amdgcn_wmma_f32_16x16x4_f32(false, A4[k], false, bm, (short)0, c, false, false);
    }
#pragma unroll
    for (int r = 0; r < 8; ++r) {
      int ch = r + 8 * hi;
      float v = c[r] + pb2[ch];
      hb[(size_t)ch * HW + p] = v;                // in place: pixel-local 1x1 conv
      ls[r] += v; lq[r] += v * v;
    }
  }
#pragma unroll
  for (int r = 0; r < 8; ++r) {
    atomicAdd(&lsum[r + 8 * hi], ls[r]);
    atomicAdd(&lsq[r + 8 * hi], lq[r]);
  }
  __syncthreads();
  if (tid < 16) { atomicAdd(&st[tid], lsum[tid]); atomicAdd(&st[16 + tid], lsq[tid]); }
}

// ---------------------------------------------------------------------------
// Kernel 5: 3x3 stride-3 zero-pad-1 conv, 16->16 ch.  aff!=null (level 0 only)
// applies BN2 affine + LReLU on load (prep is virtual).
// ---------------------------------------------------------------------------
__global__ void k_pyr(const float* __restrict__ in, float* __restrict__ out,
                      const float* __restrict__ dw, int inS, int outS,
                      const float* __restrict__ aff) {
  int idx = blockIdx.x * 256 + threadIdx.x;
  int total = NB * 16 * outS * outS;
  if (idx >= total) return;
  int ox = idx % outS;
  int tmp = idx / outS;
  int oy = tmp % outS; tmp /= outS;
  int oc = tmp & 15;
  int b = tmp >> 4;
  const float* ib = in + (size_t)b * 16 * inS * inS;
  float acc = 0.0f;
  for (int ky = 0; ky < 3; ++ky) {
    int iy = 3 * oy - 1 + ky;
    if ((unsigned)iy >= (unsigned)inS) continue;
    for (int kx = 0; kx < 3; ++kx) {
      int ix = 3 * ox - 1 + kx;
      if ((unsigned)ix >= (unsigned)inS) continue;
#pragma unroll
      for (int ic = 0; ic < 16; ++ic) {
        float v = ib[((size_t)ic * inS + iy) * inS + ix];
        if (aff) v = lrelu(v * aff[ic] + aff[16 + ic]);
        acc += dw[((oc * 16 + ic) * 3 + ky) * 3 + kx] * v;
      }
    }
  }
  out[((size_t)(b * 16 + oc) * outS + oy) * outS + ox] = acc;
}

// ---------------------------------------------------------------------------
// Kernels 6/7: all-6-iteration grouped-conv stats / final score.
// Thread (g,j): one group g, 16-pixel x-run j; 64 pixels per thread.
// z via nearest-index math directly from the tiny pyramid levels (L2 resident).
// ---------------------------------------------------------------------------
__global__ void k_stats_iter(const float* __restrict__ h1, const float* __restrict__ aff2,
                             const float* __restrict__ Db, const float* __restrict__ sw,
                             float* __restrict__ st) {
  __shared__ float lsum[192], lsq[192];
  const int tid = threadIdx.x;
  for (int i = tid; i < 192; i += 256) { lsum[i] = 0.0f; lsq[i] = 0.0f; }
  __syncthreads();

  const int g = tid >> 4, j = tid & 15;
  const float w00 = sw[g * 4 + 0], w01 = sw[g * 4 + 1];
  const float w10 = sw[g * 4 + 2], w11 = sw[g * 4 + 3];
  const float sc2 = aff2[g], sh2 = aff2[16 + g];
  const int ldim[6] = {171, 57, 19, 7, 3, 1};
  const int loff[6] = {0, 3742848, 4158720, 4204928, 4211200, 4212352};

  float s[12], q[12];
#pragma unroll
  for (int i = 0; i < 12; ++i) { s[i] = 0.0f; q[i] = 0.0f; }

  for (int t = 0; t < 64; ++t) {
    int pix = blockIdx.x * 1024 + t * 16 + j;
    int b = pix >> 18;
    int rem = pix & (HW - 1);
    int y = rem >> 9, xx = rem & 511;
    const float* hp = &h1[(size_t)(b * 16 + g) * HW + rem];
    __builtin_prefetch(hp + 1024, 0, 1);
    float prev = lrelu(hp[0] * sc2 + sh2);
#pragma unroll
    for (int lev = 0; lev < 6; ++lev) {
      int sd = ldim[lev];
      int my = (y * sd) >> 9, mx = (xx * sd) >> 9;
      float z = Db[loff[lev] + ((size_t)(b * 16 + g) * sd + my) * sd + mx];
      float v0 = w00 * prev + w01 * z;
      float v1 = w10 * prev + w11 * z;
      s[2 * lev] += v0;     q[2 * lev] += v0 * v0;
      s[2 * lev + 1] += v1; q[2 * lev + 1] += v1 * v1;
      prev = z;
    }
  }
#pragma unroll
  for (int lev = 0; lev < 6; ++lev) {
    atomicAdd(&lsum[lev * 32 + 2 * g], s[2 * lev]);
    atomicAdd(&lsq[lev * 32 + 2 * g], q[2 * lev]);
    atomicAdd(&lsum[lev * 32 + 2 * g + 1], s[2 * lev + 1]);
    atomicAdd(&lsq[lev * 32 + 2 * g + 1], q[2 * lev + 1]);
  }
  __syncthreads();
  if (tid < 192) { atomicAdd(&st[tid], lsum[tid]); atomicAdd(&st[192 + tid], lsq[tid]); }
}

__global__ void k_score(const float* __restrict__ h1, const float* __restrict__ aff2,
                        const float* __restrict__ Db, const float* __restrict__ sw,
                        const float* __restrict__ affI, float* __restrict__ out) {
  const int tid = threadIdx.x;
  const int g = tid >> 4, j = tid & 15;
  const float w00 = sw[g * 4 + 0], w01 = sw[g * 4 + 1];
  const float w10 = sw[g * 4 + 2], w11 = sw[g * 4 + 3];
  const float sc2 = aff2[g], sh2 = aff2[16 + g];
  const int ldim[6] = {171, 57, 19, 7, 3, 1};
  const int loff[6] = {0, 3742848, 4158720, 4204928, 4211200, 4212352};

  float sA[6], hA[6], sB[6], hB[6];
#pragma unroll
  for (int lev = 0; lev < 6; ++lev) {
    sA[lev] = affI[lev * 32 + 2 * g];
    hA[lev] = affI[192 + lev * 32 + 2 * g];
    sB[lev] = affI[lev * 32 + 2 * g + 1];
    hB[lev] = affI[192 + lev * 32 + 2 * g + 1];
  }

  for (int t = 0; t < 64; ++t) {
    int pix = blockIdx.x * 1024 + t * 16 + j;
    int b = pix >> 18;
    int rem = pix & (HW - 1);
    int y = rem >> 9, xx = rem & 511;
    float prev = lrelu(h1[(size_t)(b * 16 + g) * HW + rem] * sc2 + sh2);
    float acc0 = 0.0f, acc1 = 0.0f;
#pragma unroll
    for (int lev = 0; lev < 6; ++lev) {
      int sd = ldim[lev];
      int my = (y * sd) >> 9, mx = (xx * sd) >> 9;
      float z = Db[loff[lev] + ((size_t)(b * 16 + g) * sd + my) * sd + mx];
      float v0 = w00 * prev + w01 * z;
      float v1 = w10 * prev + w11 * z;
      acc0 += lrelu(v0 * sA[lev] + hA[lev]);
      acc1 += lrelu(v1 * sB[lev] + hB[lev]);
      prev = z;
    }
    out[(size_t)(b * 32 + 2 * g) * HW + rem] = acc0;
    out[(size_t)(b * 32 + 2 * g + 1) * HW + rem] = acc1;
  }
}

// ---------------------------------------------------------------------------
// Host launch. Workspace (floats):
//   [0 .. 6291456)          xq, later reused for pyramid levels (4.21M floats)
//   [6291456 .. 39845888)   h1 (conv1 raw, then conv2 raw in place)
//   [39845888 .. +896)      stats (s1 32 | s2 32 | sIter 384) + affines
// Total ~160 MB.
// ---------------------------------------------------------------------------
extern "C" void kernel_launch(void* const* d_in, const int* in_sizes, int n_in,
                              void* d_out, int out_size, void* d_ws, size_t ws_size,
                              hipStream_t stream) {
  (void)in_sizes; (void)n_in; (void)out_size; (void)ws_size;
  const float* x   = (const float*)d_in[0];
  const float* pw1 = (const float*)d_in[1];
  const float* pb1 = (const float*)d_in[2];
  const float* g1  = (const float*)d_in[3];
  const float* be1 = (const float*)d_in[4];
  const float* pw2 = (const float*)d_in[5];
  const float* pb2 = (const float*)d_in[6];
  const float* g2  = (const float*)d_in[7];
  const float* be2 = (const float*)d_in[8];
  const float* dw  = (const float*)d_in[9];
  const float* sw  = (const float*)d_in[10];
  const float* gs  = (const float*)d_in[11];
  const float* bs  = (const float*)d_in[12];

  float* ws = (float*)d_ws;
  float* xq = ws;
  float* Db = ws;                       // pyramid levels reuse xq region
  float* h1 = ws + 6291456;
  float* st = ws + 39845888;
  float* s1 = st;
  float* s2 = st + 32;
  float* sI = st + 64;
  float* aff1 = st + 448;               // scale[16], shift[16]
  float* aff2 = st + 480;
  float* affI = st + 512;               // scale[192], shift[192]

  hipMemsetAsync(st, 0, 448 * sizeof(float), stream);

  k_mode<<<dim3(32, 32, 24), 256, 0, stream>>>(x, xq);
  k_conv1<<<2048, 256, 0, stream>>>(xq, pw1, pb1, h1, s1);
  k_finalize<<<1, 256, 0, stream>>>(s1, g1, be1, aff1, aff1 + 16, 16, 16);
  k_conv2<<<2048, 256, 0, stream>>>(h1, pw2, pb2, aff1, s2);
  k_finalize<<<1, 256, 0, stream>>>(s2, g2, be2, aff2, aff2 + 16, 16, 16);

  const int ldim[7] = {512, 171, 57, 19, 7, 3, 1};
  const int loff[6] = {0, 3742848, 4158720, 4204928, 4211200, 4212352};
  for (int l = 0; l < 6; ++l) {
    int inS = ldim[l], outS = ldim[l + 1];
    int total = NB * 16 * outS * outS;
    const float* inp = (l == 0) ? (const float*)h1 : (const float*)(Db + loff[l - 1]);
    const float* aff = (l == 0) ? (const float*)aff2 : (const float*)nullptr;
    k_pyr<<<(total + 255) / 256, 256, 0, stream>>>(inp, Db + loff[l], dw, inS, outS, aff);
  }

  k_stats_iter<<<2048, 256, 0, stream>>>(h1, aff2, Db, sw, sI);
  k_finalize<<<1, 256, 0, stream>>>(sI, gs, bs, affI, affI + 192, 192, 32);
  k_score<<<2048, 256, 0, stream>>>(h1, aff2, Db, sw, affI, (float*)d_out);
}